// BarycentricPooling_63247688401278
// MI455X (gfx1250) — compile-verified
//
#include <hip/hip_runtime.h>

// BarycentricPooling on gfx1250: N=20000 nodes, S=16, HIDDEN=128, K=64, B=256.
// One wave32 per node. Cost-matrix GEMM via V_WMMA_F32_16X16X4_F32 (full f32
// precision, matches reference). Sinkhorn fully register-resident using the
// WMMA accumulator layout; cross-lane LSE via shfl_xor. Codebook staged into
// LDS by the Tensor Data Mover (TDM) with hardware row padding.

typedef __attribute__((ext_vector_type(2))) float v2f;
typedef __attribute__((ext_vector_type(8))) float v8f;
typedef __attribute__((ext_vector_type(4))) unsigned int v4u;
typedef __attribute__((ext_vector_type(4))) int v4i;
typedef __attribute__((ext_vector_type(8))) int v8i;

#define S_DIM   16
#define D_DIM   128
#define K_DIM   64
#define ITERS   20
#define LDS_STRIDE 132   // 128 + 4 pad DWORDs: 16B-aligned rows (528B) and rows
                         // advance 4 banks -> ds_load_b64 B-fetch conflict-free

__device__ __forceinline__ void lse_combine(float& m, float& s, float m2, float s2) {
    float M = fmaxf(m, m2);
    s = s * __expf(m - M) + s2 * __expf(m2 - M);
    m = M;
}

__global__ void __launch_bounds__(256) bary_zero_kernel(float* p, int n) {
    int i = blockIdx.x * 256 + threadIdx.x;
    if (i < n) p[i] = 0.0f;
}

__global__ void __launch_bounds__(256) bary_main_kernel(
    const float* __restrict__ x,         // [N,16,128]
    const int*   __restrict__ batch_idx, // [N]
    const float* __restrict__ codebook,  // [64,128]
    const float* __restrict__ logcp,     // [64]
    float* __restrict__ sums,            // [B,64] (ws)
    float* __restrict__ counts,          // [B]    (ws)
    int n_nodes)
{
    __shared__ __align__(16) float cb[K_DIM * LDS_STRIDE];
    __shared__ float y2s[K_DIM];
    __shared__ float logbs[K_DIM];
    __shared__ float lcps[K_DIM];

    const int tid = threadIdx.x;

    if (tid < K_DIM) lcps[tid] = logcp[tid];

    // ---- TDM: DMA codebook [64 rows x 128 f32] into LDS, padding each row by
    //      4 DWORDs (pad_interval=128 DW -> code 6, pad_amount=4 DW -> code 3)
    //      so LDS row stride becomes LDS_STRIDE. Issued once, by wave 0. ----
    if (tid < 32) {
        const unsigned long long gaddr = (unsigned long long)(uintptr_t)codebook;
        const unsigned int lds_base = (unsigned int)(uintptr_t)(void*)&cb[0];

        v4u g0;
        g0[0] = 1u;                                          // count=1, user D#
        g0[1] = lds_base;                                    // lds_addr
        g0[2] = (unsigned int)(gaddr & 0xFFFFFFFFu);         // global_addr lo
        g0[3] = (unsigned int)((gaddr >> 32) & 0x01FFFFFFu)  // global_addr hi
              | (2u << 30);                                  // type=2 (image)

        v8i g1;
        g1[0] = (int)((2u << 16)      // data_size = 4 bytes
                    | (1u << 20)      // pad_enable
                    | (6u << 22)      // pad_interval: 128 DWORDs
                    | (3u << 25));    // pad_amount: 4 DWORDs
        g1[1] = (int)(128u << 16);    // tensor_dim0 = 128 (bits 79:48, lo16)
        g1[2] = (int)(64u << 16);     // tensor_dim0 hi=0 | tensor_dim1 = 64 (lo16)
        g1[3] = (int)(128u << 16);    // tensor_dim1 hi=0 | tile_dim0 = 128
        g1[4] = (int)(64u);           // tile_dim1 = 64 | tile_dim2 = 0
        g1[5] = (int)(128u);          // tensor_dim0_stride = 128 (lo32)
        g1[6] = 0;                    // stride hi | tensor_dim1_stride lo (unused, 2D)
        g1[7] = 0;

        v4i gz = {0, 0, 0, 0};
#if defined(__clang_major__) && (__clang_major__ >= 23)
        v8i gz8 = {0, 0, 0, 0, 0, 0, 0, 0};
        __builtin_amdgcn_tensor_load_to_lds(g0, g1, gz, gz, gz8, 0);
#else
        __builtin_amdgcn_tensor_load_to_lds(g0, g1, gz, gz, 0);
#endif
        __builtin_amdgcn_s_wait_tensorcnt(0);
    }
    __syncthreads();

    // ---- per-block: log_b = log softmax(prior), y2[k] = ||codebook_k||^2 ----
    if (tid < K_DIM) {
        float m = lcps[0];
        #pragma unroll 8
        for (int j = 1; j < K_DIM; ++j) m = fmaxf(m, lcps[j]);
        float s = 0.0f;
        #pragma unroll 8
        for (int j = 0; j < K_DIM; ++j) s += __expf(lcps[j] - m);
        logbs[tid] = lcps[tid] - (m + __logf(s));

        const float* row = &cb[tid * LDS_STRIDE];
        float acc = 0.0f;
        #pragma unroll 8
        for (int c = 0; c < D_DIM; ++c) { float v = row[c]; acc += v * v; }
        y2s[tid] = acc;
    }
    __syncthreads();

    const int wave = tid >> 5;
    const int lane = tid & 31;
    const int node = blockIdx.x * 8 + wave;
    if (node >= n_nodes) return;          // wave-uniform -> EXEC stays all-ones

    const int half = lane >> 4;           // 0: rows 0..7 / K 0..1 ; 1: rows 8..15 / K 2..3
    const int lm   = lane & 15;
    const float log_a = -2.7725887222397811f;   // -log(16)

    // ---- GEMM: dot[s][k] = sum_d x[s][d] * y[k][d] via WMMA f32 16x16x4 ----
    // A lane layout: M = lm, Kc = 2*half + j   -> x[node][lm][4*kk + 2*half + j]
    // B lane layout: N = lm, Kc = 2*half + j   -> cb[t*16+lm][4*kk + 2*half + j]
    // D lane layout: elem v -> row (v + 8*half), col lm
    v8f acc0 = {}, acc1 = {}, acc2 = {}, acc3 = {};
    float sumsq = 0.0f;
    const float* xrow = x + (size_t)node * (S_DIM * D_DIM) + lm * D_DIM + 2 * half;
    const float* b0 = &cb[( 0 + lm) * LDS_STRIDE + 2 * half];
    const float* b1 = &cb[(16 + lm) * LDS_STRIDE + 2 * half];
    const float* b2 = &cb[(32 + lm) * LDS_STRIDE + 2 * half];
    const float* b3 = &cb[(48 + lm) * LDS_STRIDE + 2 * half];

    #pragma unroll 8
    for (int kk = 0; kk < 32; ++kk) {
        v2f a = *(const v2f*)(xrow + 4 * kk);
        sumsq += a.x * a.x + a.y * a.y;
        v2f vb0 = *(const v2f*)(b0 + 4 * kk);
        v2f vb1 = *(const v2f*)(b1 + 4 * kk);
        v2f vb2 = *(const v2f*)(b2 + 4 * kk);
        v2f vb3 = *(const v2f*)(b3 + 4 * kk);
        acc0 = __builtin_amdgcn_wmma_f32_16x16x4_f32(false, a, false, vb0, (short)0, acc0, false, false);
        acc1 = __builtin_amdgcn_wmma_f32_16x16x4_f32(false, a, false, vb1, (short)0, acc1, false, false);
        acc2 = __builtin_amdgcn_wmma_f32_16x16x4_f32(false, a, false, vb2, (short)0, acc2, false, false);
        acc3 = __builtin_amdgcn_wmma_f32_16x16x4_f32(false, a, false, vb3, (short)0, acc3, false, false);
    }

    // ---- x2 per row: lane halves hold complementary d-subsets ----
    float x2row = sumsq + __shfl_xor(sumsq, 16, 32);   // full ||x_s||^2, s = lm
    float x2v[8];
    #pragma unroll
    for (int v = 0; v < 8; ++v) x2v[v] = __shfl(x2row, v + 8 * half, 32);

    float y2t[4], lbt[4];
    #pragma unroll
    for (int t = 0; t < 4; ++t) {
        y2t[t] = y2s[t * 16 + lm];
        lbt[t] = logbs[t * 16 + lm];
    }

    // ---- Cn = C/eps = 10 * max(x2 + y2 - 2*dot, 0), in registers [tile][row] ----
    float Cn[4][8];
    {
        v8f accs[4] = { acc0, acc1, acc2, acc3 };
        #pragma unroll
        for (int t = 0; t < 4; ++t)
            #pragma unroll
            for (int v = 0; v < 8; ++v) {
                float c = fmaxf(x2v[v] + y2t[t] - 2.0f * accs[t][v], 0.0f);
                Cn[t][v] = c * 10.0f;   // 1/EPS
            }
    }

    // ---- Sinkhorn, scaled vars F=f/eps, G=g/eps; 20 scan iters + 1 extrapolation
    //      (forward-identical to 21 iterations of g-then-f) ----
    float F[8], G[4];
    #pragma unroll
    for (int v = 0; v < 8; ++v) F[v] = 0.0f;
    #pragma unroll
    for (int t = 0; t < 4; ++t) G[t] = 0.0f;

    for (int it = 0; it <= ITERS; ++it) {
        // G[k] = -( log_a + LSE_s(F_s - Cn_sk) )
        #pragma unroll
        for (int t = 0; t < 4; ++t) {
            float m = F[0] - Cn[t][0];
            #pragma unroll
            for (int v = 1; v < 8; ++v) m = fmaxf(m, F[v] - Cn[t][v]);
            float s = 0.0f;
            #pragma unroll
            for (int v = 0; v < 8; ++v) s += __expf(F[v] - Cn[t][v] - m);
            lse_combine(m, s, __shfl_xor(m, 16, 32), __shfl_xor(s, 16, 32));
            G[t] = -(log_a + m + __logf(s));
        }
        // F[s] = -LSE_k(G_k + logb_k - Cn_sk)
        #pragma unroll
        for (int v = 0; v < 8; ++v) {
            float m = G[0] + lbt[0] - Cn[0][v];
            #pragma unroll
            for (int t = 1; t < 4; ++t) m = fmaxf(m, G[t] + lbt[t] - Cn[t][v]);
            float s = 0.0f;
            #pragma unroll
            for (int t = 0; t < 4; ++t) s += __expf(G[t] + lbt[t] - Cn[t][v] - m);
            #pragma unroll
            for (int d = 1; d < 16; d <<= 1)
                lse_combine(m, s, __shfl_xor(m, d, 32), __shfl_xor(s, d, 32));
            F[v] = -(m + __logf(s));
        }
    }

    // ---- histogram: hist_k = exp( G_k + log_a + logb_k + LSE_s(F_s - Cn_sk) ) ----
    float hist[4];
    #pragma unroll
    for (int t = 0; t < 4; ++t) {
        float m = F[0] - Cn[t][0];
        #pragma unroll
        for (int v = 1; v < 8; ++v) m = fmaxf(m, F[v] - Cn[t][v]);
        float s = 0.0f;
        #pragma unroll
        for (int v = 0; v < 8; ++v) s += __expf(F[v] - Cn[t][v] - m);
        lse_combine(m, s, __shfl_xor(m, 16, 32), __shfl_xor(s, 16, 32));
        hist[t] = __expf(G[t] + log_a + lbt[t] + m + __logf(s));
    }
    float tot = hist[0] + hist[1] + hist[2] + hist[3];
    #pragma unroll
    for (int d = 1; d < 16; d <<= 1) tot += __shfl_xor(tot, d, 32);
    float inv = 1.0f / (tot + 1e-12f);

    // ---- accumulate per-graph (lanes 0..15 own unique k = t*16+lm) ----
    int b = batch_idx[node];
    if (half == 0) {
        #pragma unroll
        for (int t = 0; t < 4; ++t)
            atomicAdd(&sums[b * K_DIM + t * 16 + lm], hist[t] * inv);
        if (lm == 0) atomicAdd(&counts[b], 1.0f);
    }
}

__global__ void __launch_bounds__(256) bary_final_kernel(
    const float* __restrict__ sums, const float* __restrict__ counts,
    const float* __restrict__ logcp, float* __restrict__ out, int B)
{
    int i = blockIdx.x * 256 + threadIdx.x;
    if (i >= B * K_DIM) return;
    int b = i >> 6, k = i & 63;
    float c = counts[b];
    if (c > 0.0f) {
        out[i] = sums[i] / fmaxf(c, 1.0f);
    } else {
        // empty graph -> prior = softmax(log_codebook_prior)
        float m = logcp[0];
        #pragma unroll 8
        for (int j = 1; j < K_DIM; ++j) m = fmaxf(m, logcp[j]);
        float s = 0.0f;
        #pragma unroll 8
        for (int j = 0; j < K_DIM; ++j) s += __expf(logcp[j] - m);
        out[i] = __expf(logcp[k] - m) / s;
    }
}

extern "C" void kernel_launch(void* const* d_in, const int* in_sizes, int n_in,
                              void* d_out, int out_size, void* d_ws, size_t ws_size,
                              hipStream_t stream) {
    const float* x    = (const float*)d_in[0];   // [N,16,128] f32
    const int*   bidx = (const int*)  d_in[1];   // [N] int32
    const float* cbk  = (const float*)d_in[2];   // [64,128] f32
    const float* lcp  = (const float*)d_in[3];   // [64] f32
    // d_in[4] = num_graphs (device scalar); B derived host-side from out_size.

    const int n_nodes = in_sizes[0] / (S_DIM * D_DIM);
    const int B       = out_size / K_DIM;

    float* sums   = (float*)d_ws;                 // [B,64]
    float* counts = sums + (size_t)B * K_DIM;     // [B]
    const int zn = B * K_DIM + B;

    bary_zero_kernel<<<(zn + 255) / 256, 256, 0, stream>>>(sums, zn);
    bary_main_kernel<<<(n_nodes + 7) / 8, 256, 0, stream>>>(
        x, bidx, cbk, lcp, sums, counts, n_nodes);
    bary_final_kernel<<<(B * K_DIM + 255) / 256, 256, 0, stream>>>(
        sums, counts, lcp, (float*)d_out, B);
}